// BidirMambav2_4071628996981
// MI455X (gfx1250) — compile-verified
//
#include <hip/hip_runtime.h>

// ---------------- problem constants ----------------
#define D_MODEL   128
#define D_STATE   16
#define D_CONV    4
#define D_INNER   256
#define BSZ       4
#define SEQ       8192
#define NROWS     (BSZ * SEQ)          // 32768 (b*L rows)
#define PROJ_ROWS 288                  // 256 dt-raw + 16 B + 16 C
#define CHUNK     128
#define NCHUNK    (SEQ / CHUNK)        // 64  -> 512 scan blocks, serial depth 128

typedef float v2f __attribute__((ext_vector_type(2)));
typedef float v8f __attribute__((ext_vector_type(8)));

#if defined(__has_builtin)
#if __has_builtin(__builtin_amdgcn_wmma_f32_16x16x4_f32)
#define HAVE_WMMA_F32 1
#endif
#endif

// ---------------- fast math helpers (map to v_exp_f32 / v_log_f32) ----------------
__device__ __forceinline__ float fast_exp(float x) {          // e^x
  return __builtin_amdgcn_exp2f(x * 1.44269504f);
}
__device__ __forceinline__ float softplus_f(float x) {        // ln(1+e^x)
  if (x > 20.f) return x;
  return 0.69314718056f * __builtin_amdgcn_logf(1.f + fast_exp(x));
}
__device__ __forceinline__ float silu_f(float x) {
  return x / (1.f + fast_exp(-x));
}

// ---------------- fp32 WMMA GEMM: C[M,N] = A[M,K] @ W[N,K]^T ----------------
// One wave computes a 16 x (16*NT) strip of C: the A fragment is loaded once per
// K-step and reused across NT WMMAs. The K-loop is explicitly double-buffered:
// loads for k0+4 are issued before the WMMAs consuming k0's fragments, so the
// matrix pipe overlaps with L0/L2 latency (partial s_wait_loadcnt).
// A-frag (16x4): lane l<16 holds A[m=l][k0,k0+1], lane 16+l holds A[m=l][k0+2,k0+3]
// B-frag (4x16): lane l<16 holds B[k0][n=l],B[k0+1][n=l]; lanes 16.. hold k0+2,k0+3
// C/D: VGPR r -> (m0+r, n0+lane) lanes 0-15; (m0+8+r, n0+lane-16) lanes 16-31
template <int NT>
__global__ void gemm_xt_wmma(const float* __restrict__ A, const float* __restrict__ W,
                             float* __restrict__ C, int M, int N, int K)
{
  const int gw     = (blockIdx.x * blockDim.x + threadIdx.x) >> 5;
  const int lane   = threadIdx.x & 31;
  const int gn     = (N >> 4) / NT;            // N-tile groups
  const int groups = (M >> 4) * gn;
  if (gw >= groups) return;                    // uniform per wave -> EXEC all-ones inside
  const int m0   = (gw / gn) << 4;
  const int n0   = (gw % gn) * (16 * NT);
  const int half = lane >> 4;                  // which K-pair this lane holds
  const int mr   = lane & 15;
#ifdef HAVE_WMMA_F32
  v8f acc[NT];
#pragma unroll
  for (int i = 0; i < NT; ++i) acc[i] = (v8f){0.f, 0.f, 0.f, 0.f, 0.f, 0.f, 0.f, 0.f};
  const float* Ap = A + (size_t)(m0 + mr) * K + (half << 1);
  const float* Wp = W + (size_t)(n0 + mr) * K + (half << 1);

  // prologue: fragments for k0 = 0
  v2f a_cur = *(const v2f*)(Ap);
  v2f b_cur[NT];
#pragma unroll
  for (int i = 0; i < NT; ++i) b_cur[i] = *(const v2f*)(Wp + (size_t)i * 16 * K);

  for (int k0 = 0; k0 < K; k0 += 4) {
    // issue next k-step's loads first (clamped at the end; uniform control flow)
    const int kn = (k0 + 4 < K) ? (k0 + 4) : k0;
    v2f a_nxt = *(const v2f*)(Ap + kn);
    v2f b_nxt[NT];
#pragma unroll
    for (int i = 0; i < NT; ++i) b_nxt[i] = *(const v2f*)(Wp + (size_t)i * 16 * K + kn);

    // consume resident fragments
#pragma unroll
    for (int i = 0; i < NT; ++i)
      acc[i] = __builtin_amdgcn_wmma_f32_16x16x4_f32(false, a_cur, false, b_cur[i],
                                                     (short)0, acc[i], false, false);
    // rotate double buffers
    a_cur = a_nxt;
#pragma unroll
    for (int i = 0; i < NT; ++i) b_cur[i] = b_nxt[i];
  }
  float* Cp = C + (size_t)(m0 + (half << 3)) * N + n0 + mr;
#pragma unroll
  for (int i = 0; i < NT; ++i)
#pragma unroll
    for (int r = 0; r < 8; ++r) Cp[(size_t)r * N + i * 16] = acc[i][r];
#else
  // VALU fallback (same tiling) — only if the f32 WMMA builtin is absent
  for (int i = 0; i < NT; ++i)
    for (int r = 0; r < 8; ++r) {
      const int m = m0 + (half << 3) + r;
      const int n = n0 + i * 16 + mr;
      float s = 0.f;
      for (int k = 0; k < K; ++k) s = fmaf(A[(size_t)m * K + k], W[(size_t)n * K + k], s);
      C[(size_t)m * N + n] = s;
    }
#endif
}

// ---------------- fold dt_projs_weight @ x_proj_weight[0:8] into one 288x256 matrix ----------------
__global__ void build_wc_k(const float* __restrict__ xpw,   // (2,40,256)
                           const float* __restrict__ dtpw,  // (2,256,8)
                           float* __restrict__ Wc)          // (2,288,256)
{
  const int idx = blockIdx.x * blockDim.x + threadIdx.x;
  if (idx >= 2 * PROJ_ROWS * 256) return;
  const int ch = idx & 255;
  const int r  = (idx >> 8) % PROJ_ROWS;
  const int k  = idx / (PROJ_ROWS * 256);
  const float* xp = xpw + (size_t)k * 40 * 256;
  float v;
  if (r < 256) {                                  // dt-raw rows: dtpw[k,r,:] @ xpw[k,0:8,ch]
    const float* dw = dtpw + ((size_t)k * 256 + r) * 8;
    v = 0.f;
#pragma unroll
    for (int j = 0; j < 8; ++j) v = fmaf(dw[j], xp[j * 256 + ch], v);
  } else {                                        // B rows (8..23), C rows (24..39)
    v = xp[(8 + (r - 256)) * 256 + ch];
  }
  Wc[idx] = v;
}

// ---------------- causal depthwise conv (w=4) + bias + SiLU; time-major in/out ----------------
__global__ void conv_silu_k(const float* __restrict__ xz,   // (b*L, 512), x = cols 0..255
                            const float* __restrict__ cw,   // (256,1,4)
                            const float* __restrict__ cb,   // (256)
                            float* __restrict__ xc)         // (b*L, 256)
{
  const int row = blockIdx.x;              // b*SEQ + t
  const int t   = row & (SEQ - 1);
  const int ch  = threadIdx.x;
  const float w0 = cw[ch * 4 + 0], w1 = cw[ch * 4 + 1];
  const float w2 = cw[ch * 4 + 2], w3 = cw[ch * 4 + 3];
  const float* base = xz + (size_t)row * 512 + ch;
  float acc = cb[ch];
  if (t >= 3) acc = fmaf(w0, base[-(3 * 512)], acc);
  if (t >= 2) acc = fmaf(w1, base[-(2 * 512)], acc);
  if (t >= 1) acc = fmaf(w2, base[-(1 * 512)], acc);
  acc = fmaf(w3, base[0], acc);
  xc[(size_t)row * 256 + ch] = silu_f(acc);
}

__device__ __forceinline__ int torig_of(int k, int sidx) {
  return k ? (SEQ - 1 - sidx) : sidx;
}

// ---------------- chunked selective scan, phase A: per-chunk (prod a, local final h) ----------------
// Software-pipelined: step s+1's loads (dt/u vector; B wave-uniform -> SMEM) are
// issued before step s's dependent exp/FMA chain, hiding memory latency in the
// serial recurrence.
__global__ void scan_chunk_state(const float* __restrict__ Pbuf,   // (2, b*L, 288)
                                 const float* __restrict__ xc,     // (b*L, 256)
                                 const float* __restrict__ A_logs, // (512,16)
                                 const float* __restrict__ dt_bias,// (512)
                                 float* __restrict__ chA, float* __restrict__ chH)
{
  const int blk   = blockIdx.x;                 // b*2*NCHUNK blocks
  const int chunk = blk & (NCHUNK - 1);
  const int k     = (blk >> 6) & 1;
  const int b     = blk >> 7;
  const int ch    = threadIdx.x;                // one lane per channel
  const int d     = k * 256 + ch;
  float a2[16], h[16], ap[16];
#pragma unroll
  for (int n = 0; n < 16; ++n) {
    a2[n] = -fast_exp(A_logs[(size_t)d * 16 + n]) * 1.44269504f;  // A*log2(e)
    h[n] = 0.f;  ap[n] = 1.f;
  }
  const float bias = dt_bias[d];
  const float* P = Pbuf + (size_t)k * NROWS * PROJ_ROWS;
  const int s0 = chunk * CHUNK;

  // prologue: load step 0
  size_t row = (size_t)b * SEQ + torig_of(k, s0);
  const float* Prow = P + row * PROJ_ROWS;
  float dtraw = Prow[ch];
  float u     = xc[row * 256 + ch];
  float Bv[16];
#pragma unroll
  for (int n = 0; n < 16; ++n) Bv[n] = Prow[256 + n];

  for (int s = 0; s < CHUNK; ++s) {
    // issue next step's loads first (clamped at chunk end; uniform control flow)
    const int snext = (s + 1 < CHUNK) ? (s + 1) : s;
    const size_t rown = (size_t)b * SEQ + torig_of(k, s0 + snext);
    const float* Pn = P + rown * PROJ_ROWS;
    const int tpf = torig_of(k, s0 + ((s + 16 < CHUNK) ? s + 16 : s));
    __builtin_prefetch(P + ((size_t)b * SEQ + tpf) * PROJ_ROWS + ch, 0, 1);
    const float dtraw_n = Pn[ch];
    const float u_n     = xc[rown * 256 + ch];
    float Bn_[16];
#pragma unroll
    for (int n = 0; n < 16; ++n) Bn_[n] = Pn[256 + n];

    // compute current step
    const float dt = softplus_f(dtraw + bias);
    const float du = dt * u;
#pragma unroll
    for (int n = 0; n < 16; ++n) {
      const float e = __builtin_amdgcn_exp2f(dt * a2[n]);  // exp(dt*A_n)
      h[n]   = fmaf(e, h[n], du * Bv[n]);
      ap[n] *= e;
    }
    // rotate double buffers
    dtraw = dtraw_n;  u = u_n;
#pragma unroll
    for (int n = 0; n < 16; ++n) Bv[n] = Bn_[n];
  }
  const size_t base = (((size_t)(b * 2 + k) * NCHUNK + chunk) * 256 + ch) * 16;
#pragma unroll
  for (int n = 0; n < 16; ++n) { chA[base + n] = ap[n]; chH[base + n] = h[n]; }
}

// ---------------- phase B: serial combine across NCHUNK chunks -> per-chunk entry state ----------------
__global__ void scan_combine(const float* __restrict__ chA, const float* __restrict__ chH,
                             float* __restrict__ Hin)
{
  const int idx = blockIdx.x * blockDim.x + threadIdx.x;   // (bk, ch, n): 8*256*16
  if (idx >= 8 * 256 * 16) return;
  const int n  = idx & 15;
  const int ch = (idx >> 4) & 255;
  const int bk = idx >> 12;
  float H = 0.f;
  for (int c = 0; c < NCHUNK; ++c) {
    const size_t o = (((size_t)bk * NCHUNK + c) * 256 + ch) * 16 + n;
    Hin[o] = H;
    H = fmaf(chA[o], H, chH[o]);
  }
}

// ---------------- phase C: replay chunk with true entry state, emit y ----------------
__global__ void scan_chunk_out(const float* __restrict__ Pbuf, const float* __restrict__ xc,
                               const float* __restrict__ A_logs, const float* __restrict__ dt_bias,
                               const float* __restrict__ Ds, const float* __restrict__ Hin,
                               float* __restrict__ y0, float* __restrict__ y1)
{
  const int blk   = blockIdx.x;
  const int chunk = blk & (NCHUNK - 1);
  const int k     = (blk >> 6) & 1;
  const int b     = blk >> 7;
  const int ch    = threadIdx.x;
  const int d     = k * 256 + ch;
  float a2[16], h[16];
  const size_t sbase = (((size_t)(b * 2 + k) * NCHUNK + chunk) * 256 + ch) * 16;
#pragma unroll
  for (int n = 0; n < 16; ++n) {
    a2[n] = -fast_exp(A_logs[(size_t)d * 16 + n]) * 1.44269504f;
    h[n]  = Hin[sbase + n];
  }
  const float bias = dt_bias[d];
  const float Dch  = Ds[d];
  const float* P = Pbuf + (size_t)k * NROWS * PROJ_ROWS;
  float* yk = k ? y1 : y0;
  const int s0 = chunk * CHUNK;

  // prologue: load step 0
  size_t row = (size_t)b * SEQ + torig_of(k, s0);
  const float* Prow = P + row * PROJ_ROWS;
  float dtraw = Prow[ch];
  float u     = xc[row * 256 + ch];
  float Bv[16], Cv[16];
#pragma unroll
  for (int n = 0; n < 16; ++n) { Bv[n] = Prow[256 + n]; Cv[n] = Prow[272 + n]; }
  size_t orow = row;

  for (int s = 0; s < CHUNK; ++s) {
    const int snext = (s + 1 < CHUNK) ? (s + 1) : s;
    const size_t rown = (size_t)b * SEQ + torig_of(k, s0 + snext);
    const float* Pn = P + rown * PROJ_ROWS;
    const int tpf = torig_of(k, s0 + ((s + 16 < CHUNK) ? s + 16 : s));
    __builtin_prefetch(P + ((size_t)b * SEQ + tpf) * PROJ_ROWS + ch, 0, 1);
    const float dtraw_n = Pn[ch];
    const float u_n     = xc[rown * 256 + ch];
    float Bn_[16], Cn_[16];
#pragma unroll
    for (int n = 0; n < 16; ++n) { Bn_[n] = Pn[256 + n]; Cn_[n] = Pn[272 + n]; }

    const float dt = softplus_f(dtraw + bias);
    const float du = dt * u;
    float y = Dch * u;
#pragma unroll
    for (int n = 0; n < 16; ++n) {
      const float e = __builtin_amdgcn_exp2f(dt * a2[n]);
      h[n] = fmaf(e, h[n], du * Bv[n]);
      y    = fmaf(h[n], Cv[n], y);
    }
    yk[orow * 256 + ch] = y;       // direction-1 lands back at original index (the flip)

    dtraw = dtraw_n;  u = u_n;  orow = rown;
#pragma unroll
    for (int n = 0; n < 16; ++n) { Bv[n] = Bn_[n]; Cv[n] = Cn_[n]; }
  }
}

// ---------------- fw+bw sum, RMSNorm, SiLU gate (in-place into y0 as Yg) ----------------
__global__ void norm_gate_k(const float* __restrict__ y0, const float* __restrict__ y1,
                            const float* __restrict__ xz, const float* __restrict__ nw,
                            float* __restrict__ Yg)
{
  __shared__ float red[256];
  const int row = blockIdx.x;
  const int ch  = threadIdx.x;
  const size_t o = (size_t)row * 256 + ch;
  const float y = y0[o] + y1[o];
  red[ch] = y * y;
  __syncthreads();
  for (int s = 128; s > 0; s >>= 1) {
    if (ch < s) red[ch] += red[ch + s];
    __syncthreads();
  }
  const float rms = rsqrtf(red[0] * (1.f / 256.f) + 1e-5f);
  const float z = xz[(size_t)row * 512 + 256 + ch];
  Yg[o] = y * rms * nw[ch] * silu_f(z);
}

// ---------------- host-side launch sequence ----------------
extern "C" void kernel_launch(void* const* d_in, const int* in_sizes, int n_in,
                              void* d_out, int out_size, void* d_ws, size_t ws_size,
                              hipStream_t stream)
{
  const float* hidden = (const float*)d_in[0];
  const float* W_in   = (const float*)d_in[1];
  const float* conv_w = (const float*)d_in[2];
  const float* conv_b = (const float*)d_in[3];
  const float* xpw    = (const float*)d_in[4];
  const float* dtpw   = (const float*)d_in[5];
  const float* dtpb   = (const float*)d_in[6];
  const float* A_logs = (const float*)d_in[7];
  const float* Ds     = (const float*)d_in[8];
  const float* nw     = (const float*)d_in[9];
  const float* W_out  = (const float*)d_in[10];
  float* out = (float*)d_out;

  float* ws = (float*)d_ws;
  size_t off = 0;
  float* xz  = ws + off; off += (size_t)NROWS * 512;              // in-proj output (x | z)
  float* xc  = ws + off; off += (size_t)NROWS * 256;              // conv+silu output, time-major
  float* Wc  = ws + off; off += (size_t)2 * PROJ_ROWS * 256;      // folded projection weights
  float* P   = ws + off; off += (size_t)2 * NROWS * PROJ_ROWS;    // dt-raw|B|C per (k,row)
  float* chA = ws + off; off += (size_t)8 * NCHUNK * 256 * 16;
  float* chH = ws + off; off += (size_t)8 * NCHUNK * 256 * 16;
  float* Hin = ws + off; off += (size_t)8 * NCHUNK * 256 * 16;
  float* y0  = ws + off; off += (size_t)NROWS * 256;
  float* y1  = ws + off; off += (size_t)NROWS * 256;

  // 1) fold dt-projection into x-projection (exact, linear algebra identity)
  build_wc_k<<<(2 * PROJ_ROWS * 256 + 255) / 256, 256, 0, stream>>>(xpw, dtpw, Wc);

  // 2) in-projection GEMM: xz = hidden @ W_in^T  (32768x512x128), 4-wide N strips
  {
    const int groups = (NROWS / 16) * ((512 / 16) / 4);
    gemm_xt_wmma<4><<<(groups * 32 + 255) / 256, 256, 0, stream>>>(hidden, W_in, xz,
                                                                   NROWS, 512, 128);
  }

  // 3) causal depthwise conv + SiLU
  conv_silu_k<<<NROWS, 256, 0, stream>>>(xz, conv_w, conv_b, xc);

  // 4) per-direction projection GEMM: P_k = xc @ Wc_k^T  (32768x288x256), 3-wide strips
  for (int k = 0; k < 2; ++k) {
    const int groups = (NROWS / 16) * ((PROJ_ROWS / 16) / 3);
    gemm_xt_wmma<3><<<(groups * 32 + 255) / 256, 256, 0, stream>>>(
        xc, Wc + (size_t)k * PROJ_ROWS * 256,
        P + (size_t)k * NROWS * PROJ_ROWS, NROWS, PROJ_ROWS, 256);
  }

  // 5) chunked bidirectional selective scan (serial depth 128+64+128 vs 8192)
  scan_chunk_state<<<BSZ * 2 * NCHUNK, 256, 0, stream>>>(P, xc, A_logs, dtpb, chA, chH);
  scan_combine<<<(8 * 256 * 16 + 255) / 256, 256, 0, stream>>>(chA, chH, Hin);
  scan_chunk_out<<<BSZ * 2 * NCHUNK, 256, 0, stream>>>(P, xc, A_logs, dtpb, Ds, Hin, y0, y1);

  // 6) fw+bw sum, RMSNorm, SiLU gate (in-place: y0 becomes gated Y)
  norm_gate_k<<<NROWS, 256, 0, stream>>>(y0, y1, xz, nw, y0);

  // 7) out-projection GEMM: out = Yg @ W_out^T  (32768x128x256), 4-wide strips
  {
    const int groups = (NROWS / 16) * ((128 / 16) / 4);
    gemm_xt_wmma<4><<<(groups * 32 + 255) / 256, 256, 0, stream>>>(y0, W_out, out,
                                                                   NROWS, 128, 256);
  }

  (void)in_sizes; (void)n_in; (void)out_size; (void)ws_size;
}